// RecommendationModel_26113401160013
// MI455X (gfx1250) — compile-verified
//
#include <hip/hip_runtime.h>

typedef __attribute__((ext_vector_type(2))) float v2f;
typedef __attribute__((ext_vector_type(8))) float v8f;

#define NU_ 100000
#define NA_ 20000
#define NE_ 50000
#define EE_ 500000

// ---------------- degree count (shared across both layers) ----------------
__global__ void count_kernel(const int* __restrict__ dst, int* __restrict__ cnt, int n) {
  int t = blockIdx.x * blockDim.x + threadIdx.x;
  if (t < n) atomicAdd(&cnt[dst[t]], 1);
}

// ---------------- edge gather + scatter-add (segment sum), 64 feats ----------------
__global__ __launch_bounds__(256) void scatter64_kernel(const float* __restrict__ H,
                                                        const int* __restrict__ src,
                                                        const int* __restrict__ dst,
                                                        float* __restrict__ S, int nedge) {
  int t = blockIdx.x * blockDim.x + threadIdx.x;
  if (t >= nedge * 16) return;
  int e = t >> 4;
  int q = (t & 15) << 2;            // float4 chunk within the 64-float row
  int s = src[e], d = dst[e];
  const float4 v = *(const float4*)(H + (size_t)s * 64 + q);
  float* p = S + (size_t)d * 64 + q;
  unsafeAtomicAdd(p + 0, v.x);      // global_atomic_add_f32, L2-resident on MI455X
  unsafeAtomicAdd(p + 1, v.y);
  unsafeAtomicAdd(p + 2, v.z);
  unsafeAtomicAdd(p + 3, v.w);
}

// ---------------- fp32 WMMA GEMM tile: C[16xN] (+)= op(A[16xK]) @ B[KxN] ----------------
// One block = one 16-row stripe; wave w handles columns [16w,16w+16).
// SCALE: multiply A rows by 1/max(cnt,1) (segment mean). LOADC: accumulate into C.
template <int K, int N, int RELU, int BIAS, int SCALE, int LOADC>
__global__ void gemm16(const float* __restrict__ A, const float* __restrict__ B,
                       float* __restrict__ C, const float* __restrict__ bias,
                       const int* __restrict__ cnt) {
  const int wave = threadIdx.x >> 5;
  const int lane = threadIdx.x & 31;
  const int m0 = blockIdx.x << 4;
  const int n0 = wave << 4;
  const int l15 = lane & 15;
  const int half = lane >> 4;
  const int row = m0 + l15;   // A-matrix row held by this lane (ISA 16x4 f32 layout)
  const int col = n0 + l15;   // B/C column held by this lane
  const int koff = half * 2;  // lanes 16-31 hold K=2,3 of each K-quad

  float scale = 1.0f;
  if (SCALE) {
    int c = cnt[row];
    scale = 1.0f / (float)(c > 1 ? c : 1);
  }

  v8f acc;
  if (LOADC) {
#pragma unroll
    for (int r = 0; r < 8; ++r) acc[r] = C[(size_t)(m0 + r + half * 8) * N + col];
  } else {
#pragma unroll
    for (int r = 0; r < 8; ++r) acc[r] = 0.0f;
  }

#pragma unroll
  for (int kk = 0; kk < K; kk += 4) {
    v2f a, bb;
    a[0] = A[(size_t)row * K + kk + koff];
    a[1] = A[(size_t)row * K + kk + koff + 1];
    if (SCALE) { a[0] *= scale; a[1] *= scale; }
    bb[0] = B[(size_t)(kk + koff) * N + col];
    bb[1] = B[(size_t)(kk + koff + 1) * N + col];
    // v_wmma_f32_16x16x4_f32  (neg_a, A, neg_b, B, c_mod, C, reuse_a, reuse_b)
    acc = __builtin_amdgcn_wmma_f32_16x16x4_f32(false, a, false, bb, (short)0, acc,
                                                false, false);
  }

  float bv = 0.0f;
  if (BIAS) bv = bias[col];
#pragma unroll
  for (int r = 0; r < 8; ++r) {
    float v = acc[r] + bv;
    if (RELU) v = fmaxf(v, 0.0f);
    C[(size_t)(m0 + r + half * 8) * N + col] = v;
  }
}

// ---------------- row LayerNorm (+optional ReLU), one wave32 per row ----------------
template <int DN, int RELU>
__global__ void layernorm_kernel(float* __restrict__ X, const float* __restrict__ g,
                                 const float* __restrict__ b, int nrows) {
  int wid = blockIdx.x * (blockDim.x >> 5) + (threadIdx.x >> 5);
  if (wid >= nrows) return;
  int lane = threadIdx.x & 31;
  float* row = X + (size_t)wid * DN;
  float x0 = row[lane];
  float x1 = (DN == 64) ? row[lane + 32] : 0.0f;
  float s = x0 + x1;
#pragma unroll
  for (int o = 16; o > 0; o >>= 1) s += __shfl_xor(s, o, 32);
  float m = s * (1.0f / DN);
  float d0 = x0 - m;
  float d1 = (DN == 64) ? (x1 - m) : 0.0f;
  float vv = d0 * d0 + d1 * d1;
#pragma unroll
  for (int o = 16; o > 0; o >>= 1) vv += __shfl_xor(vv, o, 32);
  float r = rsqrtf(vv * (1.0f / DN) + 1e-5f);
  float y0 = d0 * r * g[lane] + b[lane];
  if (RELU) y0 = fmaxf(y0, 0.0f);
  row[lane] = y0;
  if (DN == 64) {
    float y1 = d1 * r * g[lane + 32] + b[lane + 32];
    if (RELU) y1 = fmaxf(y1, 0.0f);
    row[lane + 32] = y1;
  }
}

extern "C" void kernel_launch(void* const* d_in, const int* in_sizes, int n_in,
                              void* d_out, int out_size, void* d_ws, size_t ws_size,
                              hipStream_t stream) {
  (void)in_sizes; (void)n_in; (void)out_size; (void)ws_size;
  const int NN[3] = {NU_, NA_, NE_};  // 0=user 1=artist 2=event
  const float* x[3] = {(const float*)d_in[0], (const float*)d_in[1], (const float*)d_in[2]};

  // ETS order: ua au ae ea ue eu
  const int est[6]  = {0, 1, 1, 2, 0, 2};  // src type
  const int edt[6]  = {1, 0, 2, 1, 2, 0};  // dst type
  const int cpos[6] = {4, 1, 0, 2, 5, 3};  // position in sorted conv dict (ae,au,ea,eu,ua,ue)
  const int* esrc[6]; const int* edst[6];
  for (int i = 0; i < 6; ++i) {
    esrc[i] = (const int*)d_in[3 + 2 * i];
    edst[i] = (const int*)d_in[4 + 2 * i];
  }

  // params pytree (jax flatten, dict keys sorted): layers(conv,norm)x2, then proj.
  const int L0C = 15, L0N = 33, L1C = 39, L1N = 57, PRJ = 63;
  const int npos[3] = {2, 0, 1};  // user,artist,event -> index in sorted (artist,event,user)

  // ---- workspace carve-out ----
  float* ws = (float*)d_ws;
  size_t off = 0;
  float* h0[3]; float* h1[3];
  for (int t = 0; t < 3; ++t) { h0[t] = ws + off; off += (size_t)NN[t] * 64; }
  for (int t = 0; t < 3; ++t) { h1[t] = ws + off; off += (size_t)NN[t] * 64; }
  float* sum = ws + off; off += (size_t)NU_ * 64;   // reused per edge-type pass
  int* cntb = (int*)(ws + off);
  int* cnt[6]; size_t coff = 0;
  for (int i = 0; i < 6; ++i) { cnt[i] = cntb + coff; coff += NN[edt[i]]; }

  const int sgrid = (EE_ * 16 + 255) / 256;

  // ---- degree counts (once, reused by both layers) ----
  hipMemsetAsync(cntb, 0, coff * sizeof(int), stream);
  for (int i = 0; i < 6; ++i)
    count_kernel<<<(EE_ + 255) / 256, 256, 0, stream>>>(edst[i], cnt[i], EE_);

  // ---- input projection 128->64 + ReLU ----
  for (int t = 0; t < 3; ++t) {
    const float* W  = (const float*)d_in[PRJ + 2 * npos[t]];
    const float* bb = (const float*)d_in[PRJ + 2 * npos[t] + 1];
    gemm16<128, 64, 1, 1, 0, 0><<<NN[t] / 16, 128, 0, stream>>>(x[t], W, h0[t], bb, nullptr);
  }

  // ---- layer 0: 64 -> 64, accumulate into h1 ----
  hipMemsetAsync(h1[0], 0, (size_t)(NU_ + NA_ + NE_) * 64 * sizeof(float), stream);
  for (int i = 0; i < 6; ++i) {
    int st = est[i], dt = edt[i];
    hipMemsetAsync(sum, 0, (size_t)NN[dt] * 64 * sizeof(float), stream);
    scatter64_kernel<<<sgrid, 256, 0, stream>>>(h0[st], esrc[i], edst[i], sum, EE_);
    int pb = L0C + 3 * cpos[i];
    gemm16<64, 64, 0, 1, 1, 1><<<NN[dt] / 16, 128, 0, stream>>>(
        sum, (const float*)d_in[pb], h1[dt], (const float*)d_in[pb + 1], cnt[i]);
    gemm16<64, 64, 0, 0, 0, 1><<<NN[dt] / 16, 128, 0, stream>>>(
        h0[dt], (const float*)d_in[pb + 2], h1[dt], nullptr, nullptr);
  }
  for (int t = 0; t < 3; ++t) {
    const float* g  = (const float*)d_in[L0N + 2 * npos[t]];
    const float* bb = (const float*)d_in[L0N + 2 * npos[t] + 1];
    layernorm_kernel<64, 1><<<(NN[t] + 7) / 8, 256, 0, stream>>>(h1[t], g, bb, NN[t]);
  }

  // ---- layer 1: 64 -> 32, accumulate directly into d_out (user|artist|event) ----
  float* out = (float*)d_out;
  float* a1[3] = {out, out + (size_t)NU_ * 32, out + (size_t)(NU_ + NA_) * 32};
  hipMemsetAsync(out, 0, (size_t)(NU_ + NA_ + NE_) * 32 * sizeof(float), stream);
  for (int i = 0; i < 6; ++i) {
    int st = est[i], dt = edt[i];
    hipMemsetAsync(sum, 0, (size_t)NN[dt] * 64 * sizeof(float), stream);
    scatter64_kernel<<<sgrid, 256, 0, stream>>>(h1[st], esrc[i], edst[i], sum, EE_);
    int pb = L1C + 3 * cpos[i];
    gemm16<64, 32, 0, 1, 1, 1><<<NN[dt] / 16, 64, 0, stream>>>(
        sum, (const float*)d_in[pb], a1[dt], (const float*)d_in[pb + 1], cnt[i]);
    gemm16<64, 32, 0, 0, 0, 1><<<NN[dt] / 16, 64, 0, stream>>>(
        h1[dt], (const float*)d_in[pb + 2], a1[dt], nullptr, nullptr);
  }
  for (int t = 0; t < 3; ++t) {
    const float* g  = (const float*)d_in[L1N + 2 * npos[t]];
    const float* bb = (const float*)d_in[L1N + 2 * npos[t] + 1];
    layernorm_kernel<32, 0><<<(NN[t] + 7) / 8, 256, 0, stream>>>(a1[t], g, bb, NN[t]);
  }
}